// DwtFrequencyExtractor_20572893348699
// MI455X (gfx1250) — compile-verified
//
#include <hip/hip_runtime.h>

// ---------------------------------------------------------------------------
// DWT frequency extractor for MI455X (gfx1250), wave32.
// Vertical analysis + both synthesis stages expressed as banded-filter f32 GEMMs on V_WMMA_F32_16X16X4_F32.
// ---------------------------------------------------------------------------

typedef __attribute__((ext_vector_type(2))) float v2f;
typedef __attribute__((ext_vector_type(8))) float v8f;

#define HH 512
#define WW 512
#define HO 256
#define WO 256
#define NPLANE 96   // 32 batches * 3 channels

// DB4 lowpass h, highpass g[n] = (-1)^n h[7-n]
__constant__ float c_h[8] = {
    0.2303778133088964f,  0.7148465705529155f,  0.6308807679298587f,
   -0.027983769416859854f,-0.18703481171888114f, 0.030841381835986965f,
    0.032883011666982945f,-0.010597401785069032f};
__constant__ float c_g[8] = {
   -0.010597401785069032f,-0.032883011666982945f, 0.030841381835986965f,
    0.18703481171888114f, -0.027983769416859854f,-0.6308807679298587f,
    0.7148465705529155f,  -0.2303778133088964f};

// compile-time copies for fully unrolled tap loops (fold to literals)
static constexpr float kH[8] = {
    0.2303778133088964f,  0.7148465705529155f,  0.6308807679298587f,
   -0.027983769416859854f,-0.18703481171888114f, 0.030841381835986965f,
    0.032883011666982945f,-0.010597401785069032f};
static constexpr float kG[8] = {
   -0.010597401785069032f,-0.032883011666982945f, 0.030841381835986965f,
    0.18703481171888114f, -0.027983769416859854f,-0.6308807679298587f,
    0.7148465705529155f,  -0.2303778133088964f};

static __device__ __forceinline__ v8f wmma4(v2f a, v2f b, v8f c) {
  // D = A(16x4) * B(4x16) + C, fp32 exact
  return __builtin_amdgcn_wmma_f32_16x16x4_f32(false, a, false, b,
                                               (short)0, c, false, false);
}

static __device__ __forceinline__ unsigned fkey(float f) {
  // monotonic float->uint mapping for unsigned atomic min/max
  unsigned u = __float_as_uint(f);
  return (u & 0x80000000u) ? ~u : (u | 0x80000000u);
}

__global__ void init_minmax(unsigned* mm) {
  int t = threadIdx.x;
  if (t < 64) mm[t] = (t & 1) ? 0u : 0xFFFFFFFFu;  // even=min-key, odd=max-key
}

// ---------------------------------------------------------------------------
// K2: fused analysis. WG = 128 thr (4 waves), computes 16(oy) x 64(ox) coeffs.
// Horizontal 8-tap pass (VALU) -> per-wave transposed LDS tile T[n][r];
// vertical stride-2 pass as GEMM: D[m][n] = sum_r F[m][r] * T[r][n],
// F[m][r] = filt[r - 2m] (banded), K = 40 -> 10 k-steps of WMMA f32 16x16x4.
// Bands: c1 = Fh x Th (LH), c2 = Fg x Tl (HL), c3 = Fg x Th (HH). LL skipped.
// ---------------------------------------------------------------------------
__global__ __launch_bounds__(128) void dwt_analysis(
    const float* __restrict__ x, float* __restrict__ c1,
    float* __restrict__ c2, float* __restrict__ c3) {
  __shared__ float xs[40 * 134];       // input patch (zero padded)
  __shared__ float Fh[16 * 40];        // banded vertical filter matrices
  __shared__ float Fg[16 * 40];
  __shared__ float T[4 * 2 * 16 * 40]; // per-wave [lo|hi][n][r]

  const int tid  = threadIdx.x;
  const int wave = tid >> 5, lane = tid & 31;
  const int p    = blockIdx.z;
  const int oy0  = blockIdx.y * 16;
  const int ox0  = blockIdx.x * 64;
  const long plane_in = (long)p * (HH * WW);
  const int ybase = 2 * oy0 - 3, xbase = 2 * ox0 - 3;

  for (int idx = tid; idx < 40 * 134; idx += 128) {
    int r = idx / 134, c = idx - r * 134;
    int y = ybase + r, xx = xbase + c;
    float v = 0.f;
    if ((unsigned)y < HH && (unsigned)xx < WW) v = x[plane_in + y * WW + xx];
    xs[idx] = v;
  }
  for (int idx = tid; idx < 16 * 40; idx += 128) {
    int m = idx / 40, r = idx - m * 40;
    int u = r - 2 * m;
    bool ok = (u >= 0) && (u < 8);
    Fh[idx] = ok ? c_h[u] : 0.f;
    Fg[idx] = ok ? c_g[u] : 0.f;
  }
  __syncthreads();

  { // horizontal pass: Tl/Th[n][r] = sum_t h/g[t] * xs[r][2*(j)+t]
    const int n = lane & 15, rh = lane >> 4;
    float* Tl = &T[(wave * 2 + 0) * 640];
    float* Th = &T[(wave * 2 + 1) * 640];
    const int j = wave * 16 + n;
    for (int rr = 0; rr < 20; ++rr) {
      int r = rh * 20 + rr;
      const float* row = &xs[r * 134 + 2 * j];
      float lo = 0.f, hi = 0.f;
#pragma unroll
      for (int t = 0; t < 8; ++t) { float xv = row[t]; lo += kH[t] * xv; hi += kG[t] * xv; }
      Tl[n * 40 + r] = lo;
      Th[n * 40 + r] = hi;
    }
  }
  __syncthreads();

  v8f a1 = {}, a2 = {}, a3 = {};
  const int m = lane & 15;
  const int o = (lane >> 4) * 2;
  const float* Tl = &T[(wave * 2 + 0) * 640];
  const float* Th = &T[(wave * 2 + 1) * 640];
#pragma unroll
  for (int k0 = 0; k0 < 40; k0 += 4) {
    v2f aH = *(const v2f*)&Fh[m * 40 + k0 + o];
    v2f aG = *(const v2f*)&Fg[m * 40 + k0 + o];
    v2f bL = *(const v2f*)&Tl[m * 40 + k0 + o];
    v2f bH = *(const v2f*)&Th[m * 40 + k0 + o];
    a1 = wmma4(aH, bH, a1);
    a2 = wmma4(aG, bL, a2);
    a3 = wmma4(aG, bH, a3);
  }

  const long plane_out = (long)p * (HO * WO);
  const int col   = ox0 + wave * 16 + m;
  const int rbase = oy0 + ((lane >> 4) ? 8 : 0);
#pragma unroll
  for (int i = 0; i < 8; ++i) {
    long off = plane_out + (long)(rbase + i) * WO + col;
    c1[off] = a1[i];
    c2[off] = a2[i];
    c3[off] = a3[i];
  }
}

// ---------------------------------------------------------------------------
// K3: fused synthesis + min/max. WG = 128 thr, 16(y) x 64(x) output tile.
// Since tile bases are even, both synthesis stages reduce to banded GEMMs
// with FS[m][r] = filt[2r - m], K = 12:
//  stage 1 (vertical):  A = Fh x C1 + Fg x C3 ; B = Fg x C2
//  stage 2 (horizontal): recon = A x Gg^T + B x Gh^T, where G?t[n][r]=FS?[n][r]
// Stage-1 D goes through LDS to become stage-2 A-operand.
// ---------------------------------------------------------------------------
__global__ __launch_bounds__(128) void dwt_synthesis(
    const float* __restrict__ c1, const float* __restrict__ c2,
    const float* __restrict__ c3, float* __restrict__ out,
    unsigned* __restrict__ mm) {
  __shared__ float CB[3 * 40 * 12];   // coeff tiles, transposed [band][q][r]
  __shared__ float FSh[16 * 12];
  __shared__ float FSg[16 * 12];
  __shared__ float AB[4 * 2 * 16 * 16]; // per-wave stage-1 outputs
  __shared__ float sMin[128], sMax[128];

  const int tid  = threadIdx.x;
  const int wave = tid >> 5, lane = tid & 31;
  const int p    = blockIdx.z;
  const int y0   = blockIdx.y * 16;
  const int xw0  = blockIdx.x * 64;
  const int i0   = y0 / 2 - 2;    // coeff row base
  const int oxw0 = xw0 / 2 - 2;   // coeff col base
  const long plane_c = (long)p * (HO * WO);

  for (int idx = tid; idx < 3 * 40 * 12; idx += 128) {
    int a = idx / 480, rem = idx - a * 480;
    int q = rem / 12, r = rem - q * 12;
    const float* base = (a == 0) ? c1 : (a == 1) ? c2 : c3;
    int i = i0 + r, ox = oxw0 + q;
    float v = 0.f;
    if ((unsigned)i < HO && (unsigned)ox < WO) v = base[plane_c + i * WO + ox];
    CB[idx] = v;
  }
  for (int idx = tid; idx < 16 * 12; idx += 128) {
    int m2 = idx / 12, r = idx - m2 * 12;
    int u = 2 * r - m2;
    bool ok = (u >= 0) && (u < 8);
    FSh[idx] = ok ? c_h[u] : 0.f;
    FSg[idx] = ok ? c_g[u] : 0.f;
  }
  __syncthreads();

  const int m = lane & 15;
  const int o = (lane >> 4) * 2;
  const int qb = wave * 8;

  v2f fh[3], fg[3];
#pragma unroll
  for (int k = 0; k < 3; ++k) {
    fh[k] = *(const v2f*)&FSh[m * 12 + 4 * k + o];
    fg[k] = *(const v2f*)&FSg[m * 12 + 4 * k + o];
  }

  // stage 1: vertical (dilated) synthesis
  v8f accA = {}, accB = {};
#pragma unroll
  for (int k = 0; k < 3; ++k) {
    int ko = 4 * k + o;
    v2f b1 = *(const v2f*)&CB[0 * 480 + (qb + m) * 12 + ko];
    v2f b2 = *(const v2f*)&CB[1 * 480 + (qb + m) * 12 + ko];
    v2f b3 = *(const v2f*)&CB[2 * 480 + (qb + m) * 12 + ko];
    accA = wmma4(fh[k], b1, accA);
    accA = wmma4(fg[k], b3, accA);
    accB = wmma4(fg[k], b2, accB);
  }

  float* Al = &AB[(wave * 2 + 0) * 256];
  float* Bl = &AB[(wave * 2 + 1) * 256];
  const int rb = (lane >> 4) ? 8 : 0;
#pragma unroll
  for (int i = 0; i < 8; ++i) {
    Al[(rb + i) * 16 + m] = accA[i];
    Bl[(rb + i) * 16 + m] = accB[i];
  }
  __syncthreads();

  // stage 2: horizontal (dilated) synthesis; B operands reuse fh/fg fragments
  v8f accR = {};
#pragma unroll
  for (int k = 0; k < 3; ++k) {
    int ko = 4 * k + o;
    v2f aA = *(const v2f*)&Al[m * 16 + ko];
    v2f aB = *(const v2f*)&Bl[m * 16 + ko];
    accR = wmma4(aA, fg[k], accR);
    accR = wmma4(aB, fh[k], accR);
  }

  const long plane_o = (long)p * (HH * WW);
  const int xx = xw0 + wave * 16 + m;
  float lmn = 3.4e38f, lmx = -3.4e38f;
#pragma unroll
  for (int i = 0; i < 8; ++i) {
    float v = accR[i];
    out[plane_o + (long)(y0 + rb + i) * WW + xx] = v;
    lmn = fminf(lmn, v);
    lmx = fmaxf(lmx, v);
  }
  sMin[tid] = lmn;
  sMax[tid] = lmx;
  __syncthreads();
  for (int s = 64; s > 0; s >>= 1) {
    if (tid < s) {
      sMin[tid] = fminf(sMin[tid], sMin[tid + s]);
      sMax[tid] = fmaxf(sMax[tid], sMax[tid + s]);
    }
    __syncthreads();
  }
  if (tid == 0) {
    int b = p / 3;
    atomicMin(&mm[b * 2 + 0], fkey(sMin[0]));
    atomicMax(&mm[b * 2 + 1], fkey(sMax[0]));
  }
}

__global__ void dwt_normalize(float* __restrict__ out,
                              const unsigned* __restrict__ mm, int n) {
  int idx = blockIdx.x * blockDim.x + threadIdx.x;
  if (idx >= n) return;
  int b = idx / (3 * HH * WW);
  unsigned kmn = mm[b * 2 + 0], kmx = mm[b * 2 + 1];
  float mn = __uint_as_float((kmn & 0x80000000u) ? (kmn ^ 0x80000000u) : ~kmn);
  float mx = __uint_as_float((kmx & 0x80000000u) ? (kmx ^ 0x80000000u) : ~kmx);
  out[idx] = (out[idx] - mn) / (mx - mn + 1e-8f);
}

extern "C" void kernel_launch(void* const* d_in, const int* in_sizes, int n_in,
                              void* d_out, int out_size, void* d_ws, size_t ws_size,
                              hipStream_t stream) {
  const float* x = (const float*)d_in[0];
  float* out = (float*)d_out;

  // workspace: 3 coeff bands (75.5 MB) + 64 min/max keys
  const long band = (long)NPLANE * HO * WO;
  float* c1 = (float*)d_ws;
  float* c2 = c1 + band;
  float* c3 = c2 + band;
  unsigned* mm = (unsigned*)(c3 + band);

  init_minmax<<<1, 64, 0, stream>>>(mm);

  dim3 gA(WO / 64, HO / 16, NPLANE);   // 4 x 16 x 96
  dwt_analysis<<<gA, 128, 0, stream>>>(x, c1, c2, c3);

  dim3 gS(WW / 64, HH / 16, NPLANE);   // 8 x 32 x 96
  dwt_synthesis<<<gS, 128, 0, stream>>>(c1, c2, c3, out, mm);

  int n = NPLANE * HH * WW;
  dwt_normalize<<<(n + 255) / 256, 256, 0, stream>>>(out, mm, n);
}